// RoiPoolingCaffe_5970004541864
// MI455X (gfx1250) — compile-verified
//
#include <hip/hip_runtime.h>
#include <math.h>

// ROI max pooling (Caffe semantics), tuned for MI455X (gfx1250):
//  - wave32-native block of 128 threads = 4 waves; each thread owns 4 channels
//    -> one global_load_b128 per (h,w) pixel per thread, wave-coalesced 512B.
//  - ROI math is block-uniform -> scalarized to SALU + s_load (verified).
//  - feat (8 MB) is L2-resident (192 MB L2) and reused ~300x across rois/bins:
//    keep it temporal; prefetch the next window row (global_prefetch_b8).
//  - output (51.4 MB) is write-once streamed: non-temporal global_store_b128.
//  - inner w-loop unrolled 4x with a pairwise max tree: 4 independent b128
//    loads in flight per iteration, accumulator chain length halved.

typedef float v4f __attribute__((ext_vector_type(4)));

#define FH 64
#define FW 64
#define FC 512
#define PH 7
#define PW 7
#define SSCALE 0.0625f

static __device__ __forceinline__ v4f vmax4(v4f a, v4f b) {
    v4f r;
    r.x = fmaxf(a.x, b.x);
    r.y = fmaxf(a.y, b.y);
    r.z = fmaxf(a.z, b.z);
    r.w = fmaxf(a.w, b.w);
    return r;
}

__global__ __launch_bounds__(128) void roi_pool_caffe_kernel(
    const float* __restrict__ feat,
    const float* __restrict__ rois,
    float* __restrict__ out)
{
    const int bin = blockIdx.x;            // 0..48  (ph*7+pw)
    const int n   = blockIdx.y;            // roi index
    const int ph  = bin / PW;
    const int pw  = bin - ph * PW;
    const int tid = threadIdx.x;           // 0..127 -> channels 4*tid..4*tid+3

    // Block-uniform ROI parameters (compiler scalarizes: SGPR loads + SALU).
    const float* roi = rois + n * 5;
    const int b  = (int)roi[0];
    // jnp.round == round-half-to-even -> rintf (default RNE mode)
    const int sw = (int)rintf(roi[1] * SSCALE);
    const int sh = (int)rintf(roi[2] * SSCALE);
    const int ew = (int)rintf(roi[3] * SSCALE);
    const int eh = (int)rintf(roi[4] * SSCALE);

    const float roi_h = (float)max(eh - sh + 1, 1);
    const float roi_w = (float)max(ew - sw + 1, 1);
    // Must be a true f32 divide to match JAX bin boundaries ULP-exactly.
    const float bh = roi_h / (float)PH;
    const float bw = roi_w / (float)PW;

    const int hstart = min(max((int)floorf((float)ph       * bh) + sh, 0), FH);
    const int hend   = min(max((int)ceilf ((float)(ph + 1) * bh) + sh, 0), FH);
    const int wstart = min(max((int)floorf((float)pw       * bw) + sw, 0), FW);
    const int wend   = min(max((int)ceilf ((float)(pw + 1) * bw) + sw, 0), FW);
    const bool empty = (hend <= hstart) || (wend <= wstart);

    const float* fbase = feat + (size_t)b * (FH * FW * FC) + tid * 4;

    const v4f ninf = { -INFINITY, -INFINITY, -INFINITY, -INFINITY };
    v4f m0 = ninf;
    v4f m1 = ninf;

    for (int h = hstart; h < hend; ++h) {
        const float* rowp = fbase + (h * FW) * FC;
        if (h + 1 < hend) {
            // Warm WGP$/L0 with the next row of this bin's window.
            __builtin_prefetch(fbase + ((h + 1) * FW + wstart) * FC, 0, 3);
        }
        int w = wstart;
        // 4 independent b128 loads in flight, pairwise max tree.
        for (; w + 3 < wend; w += 4) {
            const v4f a = *(const v4f*)(rowp + (w + 0) * FC);
            const v4f c = *(const v4f*)(rowp + (w + 1) * FC);
            const v4f d = *(const v4f*)(rowp + (w + 2) * FC);
            const v4f e = *(const v4f*)(rowp + (w + 3) * FC);
            m0 = vmax4(m0, vmax4(a, c));
            m1 = vmax4(m1, vmax4(d, e));
        }
        if (w + 1 < wend) {          // 2-wide remainder
            const v4f a = *(const v4f*)(rowp + (w + 0) * FC);
            const v4f c = *(const v4f*)(rowp + (w + 1) * FC);
            m0 = vmax4(m0, a);
            m1 = vmax4(m1, c);
            w += 2;
        }
        if (w < wend) {              // 1-wide remainder
            const v4f a = *(const v4f*)(rowp + w * FC);
            m0 = vmax4(m0, a);
        }
    }

    v4f m = vmax4(m0, m1);
    if (empty) {
        m = (v4f){0.0f, 0.0f, 0.0f, 0.0f};
    }

    // Streaming write-once output: non-temporal b128 store (TH=NT).
    v4f* op = (v4f*)(out + ((size_t)n * (PH * PW) + bin) * FC + tid * 4);
    __builtin_nontemporal_store(m, op);
}

extern "C" void kernel_launch(void* const* d_in, const int* in_sizes, int n_in,
                              void* d_out, int out_size, void* d_ws, size_t ws_size,
                              hipStream_t stream)
{
    const float* feat = (const float*)d_in[0];   // (1, 64, 64, 512) f32
    const float* rois = (const float*)d_in[1];   // (N, 5) f32
    float* out = (float*)d_out;                  // (N, 7, 7, 512) f32

    const int n_rois = in_sizes[1] / 5;          // 512

    dim3 grid(PH * PW, n_rois, 1);               // consecutive x-blocks share a roi
    dim3 block(128, 1, 1);                       // 4 waves (wave32), 4 ch/thread
    roi_pool_caffe_kernel<<<grid, block, 0, stream>>>(feat, rois, out);
    (void)n_in; (void)out_size; (void)d_ws; (void)ws_size;
}